// GCN_V1_16020228014637
// MI455X (gfx1250) — compile-verified
//
#include <hip/hip_runtime.h>
#include <hip/hip_bf16.h>

typedef float v2f __attribute__((ext_vector_type(2)));
typedef float v8f __attribute__((ext_vector_type(8)));

#define HID 128
#define OUTF 64

// ---------------- utility ----------------
__global__ void zero_f32_kernel(float* __restrict__ p, long long n) {
  long long t = (long long)blockIdx.x * blockDim.x + threadIdx.x;
  if (t < n) p[t] = 0.0f;
}

// deg[dst] += 1 per edge
__global__ void deg_kernel(const long long* __restrict__ dst, float* __restrict__ deg, int e) {
  int t = blockIdx.x * blockDim.x + threadIdx.x;
  if (t < e) unsafeAtomicAdd(&deg[dst[t]], 1.0f);
}

// deg -> dinv = (deg+1)^-0.5   (in place)
__global__ void dinv_kernel(float* __restrict__ deg, int n) {
  int t = blockIdx.x * blockDim.x + threadIdx.x;
  if (t < n) deg[t] = rsqrtf(deg[t] + 1.0f);
}

// ---------------- WMMA GEMM: H[N x 128] = X[N x 128] @ W[128 x 128] ----------------
// Block = 256 threads = 8 waves; each wave computes a 16-row x 128-col strip with
// V_WMMA_F32_16X16X4_F32. W is staged through LDS in a K-pair-interleaved, bank-skewed
// layout so each B operand fetch is one conflict-free ds_load_b64.
//   lds index (floats): kp*288 + col*2 + r   where K = 2*kp + r
//   (288 % 64 == 32 -> consecutive kp rows land 32 banks apart, so the two
//    half-waves (kp and kp+1) hit disjoint bank ranges)
#define KP_STRIDE 288
#define KHALF_ROWS 64   // K rows staged per phase
__global__ __launch_bounds__(256) void gemm_wmma_f32(const float* __restrict__ X,
                                                     const float* __restrict__ W,
                                                     float* __restrict__ H, int nrows) {
  __shared__ float Wlds[(KHALF_ROWS / 2) * KP_STRIDE];  // 32*288 floats = 36 KB

  const int waveInBlk = threadIdx.x >> 5;
  const int lane      = threadIdx.x & 31;
  const int row0      = (blockIdx.x * 8 + waveInBlk) * 16;
  const bool active   = (row0 < nrows);  // wave-uniform

  const int half  = lane >> 4;   // 0: lanes 0-15, 1: lanes 16-31
  const int sub   = lane & 15;
  const int arow  = row0 + sub;  // A: lane holds row M=sub (both halves)
  const int khalf = half * 2;    // A VGPR0/1 hold K=khalf, khalf+1

  v8f acc[8] = {};               // 8 col-tiles of 16 -> full 128 cols

  for (int p = 0; p < 2; ++p) {
    // ---- stage K rows [p*64, p*64+64) of W into LDS (all threads) ----
    for (int i = threadIdx.x; i < KHALF_ROWS * HID; i += 256) {
      const int klocal = i >> 7;       // 0..63
      const int col    = i & (HID - 1);
      Wlds[(klocal >> 1) * KP_STRIDE + col * 2 + (klocal & 1)] =
          W[(size_t)(p * KHALF_ROWS + klocal) * HID + col];
    }
    __syncthreads();

    if (active) {
      const int kend = p * KHALF_ROWS + KHALF_ROWS;
      for (int k = p * KHALF_ROWS; k < kend; k += 4) {
        // A 16x4 f32 layout: lanes 0-15 -> K=k+0,k+1 ; lanes 16-31 -> K=k+2,k+3
        const float* ap = X + (size_t)arow * HID + k + khalf;
        v2f a; a.x = ap[0]; a.y = ap[1];
        const int kpl = ((k - p * KHALF_ROWS) >> 1) + half;  // local K-pair row
#pragma unroll
        for (int cb = 0; cb < 8; ++cb) {
          const v2f b = *(const v2f*)&Wlds[kpl * KP_STRIDE + (cb * 16 + sub) * 2];
          acc[cb] = __builtin_amdgcn_wmma_f32_16x16x4_f32(
              /*neg_a=*/false, a, /*neg_b=*/false, b,
              /*c_mod=*/(short)0, acc[cb], /*reuse_a=*/false, /*reuse_b=*/false);
        }
      }
    }
    __syncthreads();  // protect LDS before restage
  }

  if (active) {
    // C/D 16x16 f32 layout: VGPR r -> M=r (lanes 0-15), M=r+8 (lanes 16-31), N=lane&15
    const int srow = row0 + half * 8;
#pragma unroll
    for (int cb = 0; cb < 8; ++cb) {
#pragma unroll
      for (int r = 0; r < 8; ++r) {
        H[(size_t)(srow + r) * HID + cb * 16 + sub] = acc[cb][r];
      }
    }
  }
}

// ---------------- edge scatter: out[dst] += h[src] * dinv[src]*dinv[dst] ----------------
// 4 features per thread: one float4 load of h[src], 4 hardware f32 atomics.
__global__ void scatter_edges_kernel(const float* __restrict__ H,
                                     const long long* __restrict__ src,
                                     const long long* __restrict__ dst,
                                     const float* __restrict__ dinv,
                                     float* __restrict__ out, long long e) {
  long long t = (long long)blockIdx.x * blockDim.x + threadIdx.x;
  if (t >= e * (HID / 4)) return;
  const int q = (int)(t & (HID / 4 - 1));  // 0..31 -> feature quad
  const long long ed = t >> 5;
  const long long s = src[ed], d = dst[ed];
  const float w = dinv[s] * dinv[d];
  const float4 hv = *(const float4*)&H[s * HID + q * 4];
  float* o = &out[d * HID + q * 4];
  unsafeAtomicAdd(o + 0, hv.x * w);
  unsafeAtomicAdd(o + 1, hv.y * w);
  unsafeAtomicAdd(o + 2, hv.z * w);
  unsafeAtomicAdd(o + 3, hv.w * w);
}

// ---------------- X = relu(X + H*dinv^2 + bias) (in place on agg buffer) ----------------
__global__ void combine_kernel(float* __restrict__ X, const float* __restrict__ H,
                               const float* __restrict__ dinv,
                               const float* __restrict__ bias, long long n) {
  long long t = (long long)blockIdx.x * blockDim.x + threadIdx.x;
  if (t >= n * HID) return;
  const int f = (int)(t & (HID - 1));
  const long long node = t >> 7;
  const float sn = dinv[node] * dinv[node];
  const float v = X[t] + H[t] * sn + bias[f];
  X[t] = fmaxf(v, 0.0f);
}

// ---------------- mean pool: pooled[batch[n]] += x[n]; cnt[batch[n]] += 1 ----------------
__global__ void pool_kernel(const float* __restrict__ X, const long long* __restrict__ batch,
                            float* __restrict__ pooled, float* __restrict__ cnt, long long n) {
  long long t = (long long)blockIdx.x * blockDim.x + threadIdx.x;
  if (t >= n * HID) return;
  const int f = (int)(t & (HID - 1));
  const long long node = t >> 7;
  const long long g = batch[node];
  unsafeAtomicAdd(&pooled[g * HID + f], X[t]);
  if (f == 0) unsafeAtomicAdd(&cnt[g], 1.0f);
}

// ---------------- out[g][o] = (pooled[g]/max(cnt,1)) @ W_out + b_out ----------------
__global__ void out_gemm_kernel(const float* __restrict__ pooled, const float* __restrict__ cnt,
                                const float* __restrict__ W_out, const float* __restrict__ b_out,
                                float* __restrict__ out) {
  const int g = blockIdx.x;      // G blocks
  const int o = threadIdx.x;     // OUTF threads
  const float inv = 1.0f / fmaxf(cnt[g], 1.0f);
  float acc = b_out[o];
  for (int k = 0; k < HID; ++k)
    acc += pooled[g * HID + k] * inv * W_out[k * OUTF + o];
  out[g * OUTF + o] = acc;
}

static inline int cdiv(long long a, long long b) { return (int)((a + b - 1) / b); }

extern "C" void kernel_launch(void* const* d_in, const int* in_sizes, int n_in,
                              void* d_out, int out_size, void* d_ws, size_t ws_size,
                              hipStream_t stream) {
  const float*     x     = (const float*)d_in[0];
  const long long* ei    = (const long long*)d_in[1];
  const long long* batch = (const long long*)d_in[2];
  const float*     Ws    = (const float*)d_in[3];
  const float*     bs    = (const float*)d_in[4];
  const float*     W_out = (const float*)d_in[5];
  const float*     b_out = (const float*)d_in[6];
  float*           out   = (float*)d_out;

  const long long N = in_sizes[0] / HID;          // 50000
  const long long E = in_sizes[1] / 2;            // 800000
  const int       L = in_sizes[3] / (HID * HID);  // 3
  const int       G = out_size / OUTF;            // 64

  const long long* src = ei;
  const long long* dst = ei + E;

  // workspace layout (floats)
  float* ws     = (float*)d_ws;
  float* dinv   = ws;                 // [N]   (used as deg first)
  float* h      = dinv + N;           // [N*HID]
  float* bufA   = h + N * HID;        // [N*HID]
  float* bufB   = bufA + N * HID;     // [N*HID]
  float* pooled = bufB + N * HID;     // [G*HID]
  float* cnt    = pooled + (long long)G * HID;  // [G]

  const int TB = 256;
  const long long NF = N * HID;
  const long long EQ = E * (HID / 4);

  // 1) degree + symmetric norm
  zero_f32_kernel<<<cdiv(N, TB), TB, 0, stream>>>(dinv, N);
  deg_kernel<<<cdiv(E, TB), TB, 0, stream>>>(dst, dinv, (int)E);
  dinv_kernel<<<cdiv(N, TB), TB, 0, stream>>>(dinv, (int)N);

  // 2) GCN layers
  const int tiles       = (int)((N + 15) / 16);      // 3125
  const int gemm_blocks = cdiv(tiles, 8);            // 8 waves (row tiles) per block
  for (int i = 0; i < L; ++i) {
    const float* Xin  = (i == 0) ? x : ((i & 1) ? bufA : bufB);
    float*       Xout = (i & 1) ? bufB : bufA;
    gemm_wmma_f32<<<gemm_blocks, TB, 0, stream>>>(Xin, Ws + (size_t)i * HID * HID, h, (int)N);
    zero_f32_kernel<<<cdiv(NF, TB), TB, 0, stream>>>(Xout, NF);
    scatter_edges_kernel<<<cdiv(EQ, TB), TB, 0, stream>>>(h, src, dst, dinv, Xout, E);
    combine_kernel<<<cdiv(NF, TB), TB, 0, stream>>>(Xout, h, dinv, bs + (size_t)i * HID, N);
  }
  float* Xfinal = ((L - 1) & 1) ? bufB : bufA;

  // 3) global mean pool + output projection
  zero_f32_kernel<<<cdiv((long long)G * HID + G, TB), TB, 0, stream>>>(pooled, (long long)G * HID + G);
  pool_kernel<<<cdiv(NF, TB), TB, 0, stream>>>(Xfinal, batch, pooled, cnt, N);
  out_gemm_kernel<<<G, OUTF, 0, stream>>>(pooled, cnt, W_out, b_out, out);
}